// ROIMask_60464549593193
// MI455X (gfx1250) — compile-verified
//
#include <hip/hip_runtime.h>
#include <hip/hip_bf16.h>
#include <stdint.h>

// Problem constants from the reference.
#define S_ 2
#define N_ 8
#define M_ 16
#define C_ 32
#define H_ 128
#define W_ 128

#define HSPLIT 4                 // split each plane's rows across 4 blocks
#define HS (H_ / HSPLIT)         // 32 rows per block
#define PLANE (H_ * W_)          // 16384 floats per (s,n,c) plane
#define TILE (HS * W_)           // 4096 floats staged per block (16 KiB)
#define TILE4 (TILE / 4)         // 1024 float4 per tile
#define THREADS 256              // 8 wave32 waves per block

typedef float f4  __attribute__((ext_vector_type(4)));
typedef int   v4i __attribute__((vector_size(16)));   // type the async builtin expects

#if defined(__gfx1250__) && __has_builtin(__builtin_amdgcn_global_load_async_to_lds_b128)
#define HAVE_ASYNC_LDS 1
#else
#define HAVE_ASYNC_LDS 0
#endif

#if defined(__gfx1250__) && __has_builtin(__builtin_amdgcn_s_wait_asynccnt)
#define WAIT_ASYNC() __builtin_amdgcn_s_wait_asynccnt(0)
#elif defined(__gfx1250__)
#define WAIT_ASYNC() asm volatile("s_wait_asynccnt 0" ::: "memory")
#else
#define WAIT_ASYNC() ((void)0)
#endif

__global__ __launch_bounds__(THREADS) void ROIMask_60464549593193_kernel(
    const float* __restrict__ fm,     // [S,N,C,H,W]
    const int* __restrict__ boxes,    // [S,N,M,4] (x1,y1,x2,y2)
    float* __restrict__ out)          // [S,N*M,C,H,W]
{
    __shared__ float lds_tile[TILE];      // 16 KiB feature tile
    __shared__ int   sbox[M_ * 4];        // this image's 16 boxes

    const int bid = blockIdx.x;
    const int hsplit = bid & (HSPLIT - 1);        // which row-quarter
    const int pc = bid >> 2;                      // 0 .. S*N*C-1
    const int cid = pc % C_;
    const int sn = pc / C_;                       // flattened (s*N + n)

    const int tid = threadIdx.x;

    const float* plane = fm + ((size_t)sn * C_ + cid) * PLANE + (size_t)hsplit * TILE;

    // Stage this image's boxes (64 ints) into LDS.
    if (tid < M_ * 4) sbox[tid] = boxes[sn * (M_ * 4) + tid];

    // ---- Stage the 32x128 feature tile into LDS ------------------------------
#if HAVE_ASYNC_LDS
    {
        // CDNA5 async global->LDS DMA path (ASYNCcnt-tracked, no VGPR staging).
        v4i* gsrc = (v4i*)plane;          // 16-byte granules in global memory
        v4i* ldst = (v4i*)(&lds_tile[0]); // 16-byte granules in LDS
#pragma unroll
        for (int i = 0; i < TILE4 / THREADS; ++i) {   // 4 x b128 per lane
            const int k = i * THREADS + tid;
            __builtin_amdgcn_global_load_async_to_lds_b128(gsrc + k, ldst + k,
                                                           /*offset=*/0, /*cpol=*/0);
        }
        WAIT_ASYNC();
    }
#else
    {
#pragma unroll
        for (int i = 0; i < TILE4 / THREADS; ++i) {
            const int idx = i * THREADS + tid;
            *(f4*)&lds_tile[idx * 4] = *(const f4*)&plane[idx * 4];
        }
    }
#endif
    __syncthreads();

    // ---- Emit 16 masked copies of the tile (pure streaming NT stores) --------
    // Loop nest: each float4 is read from LDS ONCE, then broadcast to all 16
    // box-masked output planes (stores stride by C*H*W floats = 2 MiB).
#pragma unroll
    for (int i = 0; i < TILE4 / THREADS; ++i) {
        const int idx = i * THREADS + tid;       // 0..1023
        const int hh  = idx >> 5;                // local row 0..31
        const int w4i = idx & 31;                // float4 column 0..31
        const int h   = hsplit * HS + hh;        // global row
        const int x0  = w4i * 4;

        const f4 v = *(const f4*)&lds_tile[hh * W_ + x0];

        float* obase = out + ((size_t)(sn * M_) * C_ + cid) * PLANE
                           + (size_t)hsplit * TILE + hh * W_ + x0;

        for (int m = 0; m < M_; ++m) {
            const int bx1 = sbox[m * 4 + 0];
            const int by1 = sbox[m * 4 + 1];
            const int bx2 = sbox[m * 4 + 2];
            const int by2 = sbox[m * 4 + 3];

            const bool inY = (h >= by1) && (h < by2);
            f4 r;
            r.x = (inY && (x0 + 0 >= bx1) && (x0 + 0 < bx2)) ? v.x : 0.0f;
            r.y = (inY && (x0 + 1 >= bx1) && (x0 + 1 < bx2)) ? v.y : 0.0f;
            r.z = (inY && (x0 + 2 >= bx1) && (x0 + 2 < bx2)) ? v.z : 0.0f;
            r.w = (inY && (x0 + 3 >= bx1) && (x0 + 3 < bx2)) ? v.w : 0.0f;

            // Non-temporal 128-bit store: output is write-once, never re-read;
            // keep the 32 MiB of feature maps resident in L2 instead.
            __builtin_nontemporal_store(r, (f4*)(obase + (size_t)m * (C_ * PLANE)));
        }
    }
}

extern "C" void kernel_launch(void* const* d_in, const int* in_sizes, int n_in,
                              void* d_out, int out_size, void* d_ws, size_t ws_size,
                              hipStream_t stream) {
    const float* fm    = (const float*)d_in[0];
    const int*   boxes = (const int*)d_in[1];
    float*       out   = (float*)d_out;

    const int blocks = S_ * N_ * C_ * HSPLIT;   // 2048 blocks, 8 waves each
    ROIMask_60464549593193_kernel<<<blocks, THREADS, 0, stream>>>(fm, boxes, out);
}